// dare_63067299774945
// MI455X (gfx1250) — compile-verified
//
#include <hip/hip_runtime.h>

typedef __attribute__((ext_vector_type(2))) float v2f;
typedef __attribute__((ext_vector_type(8))) float v8f;

#define LDM 33   // row stride for 32x32 LDS matrices (pad to dodge bank conflicts)
#define LDE 65   // row stride for augmented [W | I] (32 x 64)

struct Smem {
    float A [32 * LDM];
    float G [32 * LDM];
    float H [32 * LDM];
    float T1[32 * LDM];
    float T2[32 * LDM];
    float T3[32 * LDM];
    float U [32 * LDM];
    float E [32 * LDE];   // augmented matrix / scratch (2080 floats)
    float Bs[32 * 17];    // B is 32x16, stride 17
};

// One 16x16 tile (tm,tn) of a 32x32 product D = op(A)*op(B) [+ C], K=32,
// one wave32 -> 8 chained V_WMMA_F32_16X16X4_F32. f32 fragment layout (ISA 7.12.2):
//   A 16x4 : lanes 0-15 M=lane, VGPR0/1 = K0/K1 ; lanes 16-31 = K2/K3
//   B 4x16 : VGPR0 = rows K0/K2 (lane halves), VGPR1 = K1/K3 ; N = lane&15
//   C/D    : VGPR r -> M = r (lanes 0-15) / r+8 (lanes 16-31), N = lane&15
template <bool TA, bool TB, bool ACC>
__device__ __forceinline__ void mm32(float* __restrict__ D, int ldd,
                                     const float* __restrict__ A, int lda,
                                     const float* __restrict__ B, int ldb,
                                     const float* __restrict__ C, int ldc,
                                     int lane, int tm, int tn)
{
    const int half = lane >> 4;
    const int l15  = lane & 15;

    v8f c;
    if (ACC) {
#pragma unroll
        for (int r = 0; r < 8; ++r)
            c[r] = C[(tm + r + 8 * half) * ldc + tn + l15];
    } else {
#pragma unroll
        for (int r = 0; r < 8; ++r) c[r] = 0.0f;
    }

#pragma unroll
    for (int k0 = 0; k0 < 32; k0 += 4) {
        const int ka = k0 + 2 * half;
        v2f a, b;
        if (!TA) {
            a.x = A[(tm + l15) * lda + ka];
            a.y = A[(tm + l15) * lda + ka + 1];
        } else {                    // logical A[m][k] = mem[k][m]
            a.x = A[ka * lda + tm + l15];
            a.y = A[(ka + 1) * lda + tm + l15];
        }
        if (!TB) {
            b.x = B[ka * ldb + tn + l15];
            b.y = B[(ka + 1) * ldb + tn + l15];
        } else {                    // logical B[k][n] = mem[n][k]
            b.x = B[(tn + l15) * ldb + ka];
            b.y = B[(tn + l15) * ldb + ka + 1];
        }
        c = __builtin_amdgcn_wmma_f32_16x16x4_f32(false, a, false, b,
                                                  (short)0, c, false, false);
    }

#pragma unroll
    for (int r = 0; r < 8; ++r)
        D[(tm + r + 8 * half) * ldd + tn + l15] = c[r];
}

// Two products sharing the same (non-transposed) left operand: D1 = A*B1, D2 = A*B2.
__device__ __forceinline__ void mm32_pair(float* __restrict__ D1, float* __restrict__ D2,
                                          int ldd,
                                          const float* __restrict__ A, int lda,
                                          const float* __restrict__ B1,
                                          const float* __restrict__ B2, int ldb,
                                          int lane, int tm, int tn)
{
    const int half = lane >> 4;
    const int l15  = lane & 15;

    v8f c1, c2;
#pragma unroll
    for (int r = 0; r < 8; ++r) { c1[r] = 0.0f; c2[r] = 0.0f; }

#pragma unroll
    for (int k0 = 0; k0 < 32; k0 += 4) {
        const int ka = k0 + 2 * half;
        v2f a, b1, b2;
        a.x  = A[(tm + l15) * lda + ka];
        a.y  = A[(tm + l15) * lda + ka + 1];
        b1.x = B1[ka * ldb + tn + l15];
        b1.y = B1[(ka + 1) * ldb + tn + l15];
        b2.x = B2[ka * ldb + tn + l15];
        b2.y = B2[(ka + 1) * ldb + tn + l15];
        c1 = __builtin_amdgcn_wmma_f32_16x16x4_f32(false, a, false, b1,
                                                   (short)0, c1, false, false);
        c2 = __builtin_amdgcn_wmma_f32_16x16x4_f32(false, a, false, b2,
                                                   (short)0, c2, false, false);
    }

#pragma unroll
    for (int r = 0; r < 8; ++r) {
        D1[(tm + r + 8 * half) * ldd + tn + l15] = c1[r];
        D2[(tm + r + 8 * half) * ldd + tn + l15] = c2[r];
    }
}

__global__ __launch_bounds__(128) void dare_sda_kernel(
    const float* __restrict__ gA, const float* __restrict__ gB,
    const float* __restrict__ gQ, const float* __restrict__ gR,
    float* __restrict__ gP)
{
    __shared__ Smem s;
    const int b    = blockIdx.x;
    const int t    = threadIdx.x;
    const int lane = t & 31;
    const int wv   = t >> 5;
    const int tm   = (wv & 1) * 16;
    const int tn   = (wv >> 1) * 16;

    const float* Ab = gA + (size_t)b * 1024;
    const float* Bb = gB + (size_t)b * 512;
    const float* Qb = gQ + (size_t)b * 1024;
    const float* Rb = gR + (size_t)b * 256;

    // ---- load inputs into LDS (coalesced) ----
    for (int i = t; i < 1024; i += 128) s.A [(i >> 5) * LDM + (i & 31)] = Ab[i];
    for (int i = t; i < 512;  i += 128) s.Bs[(i >> 4) * 17  + (i & 15)] = Bb[i];
    for (int i = t; i < 1024; i += 128) s.T3[(i >> 5) * LDM + (i & 31)] = Qb[i]; // raw Q
    for (int i = t; i < 256;  i += 128) s.U [(i >> 4) * LDM + (i & 15)] = Rb[i]; // raw R
    __syncthreads();

    // H0 = 0.5 (Q + Q^T) ; E[:16, :32] = [0.5(R+R^T) | I]
    for (int i = t; i < 1024; i += 128) {
        int r = i >> 5, c = i & 31;
        s.H[r * LDM + c] = 0.5f * (s.T3[r * LDM + c] + s.T3[c * LDM + r]);
    }
    for (int i = t; i < 256; i += 128) {
        int r = i >> 4, c = i & 15;
        s.E[r * LDE + c]      = 0.5f * (s.U[r * LDM + c] + s.U[c * LDM + r]);
        s.E[r * LDE + 16 + c] = (r == c) ? 1.0f : 0.0f;
    }

    // ---- Gauss-Jordan, deferred normalization: Rinv = diag^-1 * E[:16,16:32] ----
    // At step k the pivot row is nonzero only in augmented cols [k+1, 16+k]:
    // 16 contiguous cols -> 2 per thread (8 threads/row), 1 barrier/step.
#pragma unroll 1
    for (int k = 0; k < 16; ++k) {
        __syncthreads();
        int r = t >> 3;                       // 0..15
        if (r != k) {
            float rinv = __builtin_amdgcn_rcpf(s.E[k * LDE + k]);
            float f    = s.E[r * LDE + k] * rinv;
            int cb = k + 1 + (t & 7) * 2;
#pragma unroll
            for (int j = 0; j < 2; ++j) {
                int c = cb + j;
                if (c < 32) s.E[r * LDE + c] -= f * s.E[k * LDE + c];
            }
        }
    }
    __syncthreads();
    for (int i = t; i < 256; i += 128) {      // scale right half by 1/diag
        int r = i >> 4, c = i & 15;
        s.E[r * LDE + 16 + c] *= __builtin_amdgcn_rcpf(s.E[r * LDE + r]);
    }
    __syncthreads();

    // T1 = B @ Rinv (32x16), then G0 = T1 @ B^T (32x32)  -- once per batch, VALU
    for (int i = t; i < 512; i += 128) {
        int r = i >> 4, c = i & 15;
        float acc = 0.0f;
#pragma unroll
        for (int k = 0; k < 16; ++k) acc += s.Bs[r * 17 + k] * s.E[k * LDE + 16 + c];
        s.T1[r * LDM + c] = acc;
    }
    __syncthreads();
    for (int i = t; i < 1024; i += 128) {
        int r = i >> 5, c = i & 31;
        float acc = 0.0f;
#pragma unroll
        for (int k = 0; k < 16; ++k) acc += s.T1[r * LDM + k] * s.Bs[c * 17 + k];
        s.G[r * LDM + c] = acc;
    }

    float* pA  = s.A;  float* pG  = s.G;  float* pH  = s.H;
    float* pT1 = s.T1; float* pT2 = s.T2; float* pT3 = s.T3;

    // ---- 30 structure-preserving doubling iterations ----
#pragma unroll 1
    for (int it = 0; it < 30; ++it) {
        __syncthreads();
        // W = G @ H  -> E[:, :32]
        mm32<false, false, false>(s.E, LDE, pG, LDM, pH, LDM, nullptr, 0, lane, tm, tn);
        __syncthreads();
        // W += I ; E[:, 32:64] = I
        if (t < 32) s.E[t * LDE + t] += 1.0f;
        for (int i = t; i < 1024; i += 128) {
            int r = i >> 5, c = i & 31;
            s.E[r * LDE + 32 + c] = (r == c) ? 1.0f : 0.0f;
        }
        // Gauss-Jordan (deferred normalization, truncated updates):
        // step k touches only augmented cols [k+1, 32+k] (36-col window, 9/thread).
#pragma unroll 1
        for (int k = 0; k < 32; ++k) {
            __syncthreads();
            int r = t >> 2;                   // 0..31
            if (r != k) {
                float rinv = __builtin_amdgcn_rcpf(s.E[k * LDE + k]);
                float f    = s.E[r * LDE + k] * rinv;
                int cb = k + 1 + (t & 3) * 9;
#pragma unroll
                for (int j = 0; j < 9; ++j) {
                    int c = cb + j;
                    if (c < 64) s.E[r * LDE + c] -= f * s.E[k * LDE + c];
                }
            }
        }
        __syncthreads();
        for (int i = t; i < 1024; i += 128) { // Winv = diag^-1 * right half
            int r = i >> 5, c = i & 31;
            s.E[r * LDE + 32 + c] *= __builtin_amdgcn_rcpf(s.E[r * LDE + r]);
        }
        __syncthreads();

        // T1 = Winv @ A ; T2 = Winv @ G   (shared A-operand fragments)
        mm32_pair(pT1, pT2, LDM, s.E + 32, LDE, pA, pG, LDM, lane, tm, tn);
        __syncthreads();
        // T3 = A @ T1  (next A) ; U = T2 @ A^T
        mm32<false, false, false>(pT3, LDM, pA, LDM, pT1, LDM, nullptr, 0, lane, tm, tn);
        mm32<false, true,  false>(s.U, LDM, pT2, LDM, pA, LDM, nullptr, 0, lane, tm, tn);
        __syncthreads();
        // nextG = G + A @ U  -> T2 ; E = H @ T1
        mm32<false, false, true >(pT2, LDM, pA, LDM, s.U, LDM, pG, LDM, lane, tm, tn);
        mm32<false, false, false>(s.E, LDE, pH, LDM, pT1, LDM, nullptr, 0, lane, tm, tn);
        __syncthreads();
        // nextH = H + A^T @ E  -> T1
        mm32<true, false, true>(pT1, LDM, pA, LDM, s.E, LDE, pH, LDM, lane, tm, tn);

        // ping-pong swap (uniform per-thread pointers)
        float* tp;
        tp = pA; pA = pT3; pT3 = tp;
        tp = pG; pG = pT2; pT2 = tp;
        tp = pH; pH = pT1; pT1 = tp;
    }
    __syncthreads();

    // P = 0.5 (H + H^T)
    float* out = gP + (size_t)b * 1024;
    for (int i = t; i < 1024; i += 128) {
        int r = i >> 5, c = i & 31;
        out[i] = 0.5f * (pH[r * LDM + c] + pH[c * LDM + r]);
    }
}

extern "C" void kernel_launch(void* const* d_in, const int* in_sizes, int n_in,
                              void* d_out, int out_size, void* d_ws, size_t ws_size,
                              hipStream_t stream)
{
    const float* A = (const float*)d_in[0];
    const float* B = (const float*)d_in[1];
    const float* Q = (const float*)d_in[2];
    const float* R = (const float*)d_in[3];
    float* P = (float*)d_out;

    const int batches = in_sizes[0] / 1024;   // A is (b, 32, 32)
    dare_sda_kernel<<<dim3(batches), dim3(128), 0, stream>>>(A, B, Q, R, P);
}